// GATEncoder_22316650070158
// MI455X (gfx1250) — compile-verified
//
#include <hip/hip_runtime.h>
#include <math.h>

#define DIN 256
#define HID 64
#define HEADS 4
#define DOUT 128
#define NEG_SLOPE 0.2f

typedef __attribute__((ext_vector_type(2))) float v2f;
typedef __attribute__((ext_vector_type(8))) float v8f;

// ---------------------------------------------------------------------------
// float atomic max via signed/unsigned integer monotone encoding
// ---------------------------------------------------------------------------
__device__ __forceinline__ void atomicMaxFloat(float* addr, float val) {
  if (val >= 0.0f)
    atomicMax((int*)addr, __float_as_int(val));
  else
    atomicMin((unsigned int*)addr, __float_as_uint(val));
}

// ---------------------------------------------------------------------------
// H[nrows x NOUT] = X[nrows x K] @ W[K x NOUT]   (fp32, V_WMMA_F32_16X16X4_F32)
// One wave per 16x64 output strip (4 WMMA column tiles sharing one A frag).
// nrows must be a multiple of 16 (N=50000 is).
// A frag 16x4: lanes 0-15 -> M=lane, K={k,k+1}; lanes 16-31 -> M=lane-16, K={k+2,k+3}
// B frag 4x16: v0 -> rows k (lanes0-15)/k+2 (lanes16-31), v1 -> k+1/k+3, N=lane%16
// C/D 16x16: VGPR v -> M=v (lanes0-15), M=v+8 (lanes16-31), N=lane%16
// ---------------------------------------------------------------------------
template<int K, int NOUT>
__global__ void gemm_f32_wmma(const float* __restrict__ X,
                              const float* __restrict__ W,
                              float* __restrict__ H, int nrows) {
  const int lane = threadIdx.x & 31;
  const int wave = threadIdx.x >> 5;
  const int colGroups = NOUT / 64;                // 4 tiles of 16 per wave
  const int strip = blockIdx.x * (blockDim.x >> 5) + wave;
  const int rowTile = strip / colGroups;
  const int colGroup = strip - rowTile * colGroups;
  if (rowTile * 16 >= nrows) return;              // wave-uniform exit
  const int r = lane & 15;
  const int half = lane >> 4;
  const float* __restrict__ Ap = X + (size_t)(rowTile * 16 + r) * K + 2 * half;
  const float* __restrict__ Bp = W + (size_t)(2 * half) * NOUT + colGroup * 64 + r;
  v8f acc0 = {}, acc1 = {}, acc2 = {}, acc3 = {};
#pragma unroll 4
  for (int k = 0; k < K; k += 4) {
    v2f a;
    a[0] = Ap[k];                                 // K = k + 2*half
    a[1] = Ap[k + 1];                             // K = k + 2*half + 1
    const float* b0 = Bp + (size_t)k * NOUT;      // row k + 2*half
    const float* b1 = Bp + (size_t)(k + 1) * NOUT;
    v2f bA, bB, bC, bD;
    bA[0] = b0[0];   bA[1] = b1[0];
    bB[0] = b0[16];  bB[1] = b1[16];
    bC[0] = b0[32];  bC[1] = b1[32];
    bD[0] = b0[48];  bD[1] = b1[48];
    acc0 = __builtin_amdgcn_wmma_f32_16x16x4_f32(false, a, false, bA,
                                                 (short)0, acc0, false, false);
    acc1 = __builtin_amdgcn_wmma_f32_16x16x4_f32(false, a, false, bB,
                                                 (short)0, acc1, false, false);
    acc2 = __builtin_amdgcn_wmma_f32_16x16x4_f32(false, a, false, bC,
                                                 (short)0, acc2, false, false);
    acc3 = __builtin_amdgcn_wmma_f32_16x16x4_f32(false, a, false, bD,
                                                 (short)0, acc3, false, false);
  }
  float* __restrict__ Cp =
      H + (size_t)(rowTile * 16 + 8 * half) * NOUT + colGroup * 64 + r;
#pragma unroll
  for (int v = 0; v < 8; ++v) {
    Cp[(size_t)v * NOUT]      = acc0[v];
    Cp[(size_t)v * NOUT + 16] = acc1[v];
    Cp[(size_t)v * NOUT + 32] = acc2[v];
    Cp[(size_t)v * NOUT + 48] = acc3[v];
  }
}

// ---------------------------------------------------------------------------
// Per-node attention coefficients: e_src[n,h] = dot(H[n,h,:], a_src[h,:])
// ---------------------------------------------------------------------------
__global__ void dot_e_kernel(const float* __restrict__ H,
                             const float* __restrict__ a_src,
                             const float* __restrict__ a_dst,
                             float* __restrict__ e_src,
                             float* __restrict__ e_dst,
                             int n, int heads, int C) {
  int idx = blockIdx.x * blockDim.x + threadIdx.x;
  if (idx >= n * heads) return;
  int node = idx / heads;
  int hd = idx - node * heads;
  const float* hp = H + (size_t)node * heads * C + (size_t)hd * C;
  const float* as = a_src + (size_t)hd * C;
  const float* ad = a_dst + (size_t)hd * C;
  float s0 = 0.f, s1 = 0.f;
  for (int c = 0; c < C; ++c) {
    float v = hp[c];
    s0 += v * as[c];
    s1 += v * ad[c];
  }
  e_src[idx] = s0;
  e_dst[idx] = s1;
}

// ---------------------------------------------------------------------------
// grid-stride fill
// ---------------------------------------------------------------------------
__global__ void fill_kernel(float* __restrict__ p, float v, size_t n) {
  size_t i = (size_t)blockIdx.x * blockDim.x + threadIdx.x;
  size_t stride = (size_t)gridDim.x * blockDim.x;
  for (; i < n; i += stride) p[i] = v;
}

// ---------------------------------------------------------------------------
// Edge pass 1: logit = leaky_relu(e_src[s]+e_dst[d]); store; segment atomic-max
// Edge id >= E encodes self-loop (src = dst = e - E), matching reference order.
// ---------------------------------------------------------------------------
__global__ void edge_logit_max(const int* __restrict__ srcIdx,
                               const int* __restrict__ dstIdx,
                               int E, int Etot,
                               const float* __restrict__ e_src,
                               const float* __restrict__ e_dst,
                               float* __restrict__ logits,
                               float* __restrict__ m, int heads) {
  int t = blockIdx.x * blockDim.x + threadIdx.x;
  if (t >= Etot * heads) return;
  int e = t / heads;
  int hd = t - e * heads;
  int s = (e < E) ? srcIdx[e] : (e - E);
  int d = (e < E) ? dstIdx[e] : (e - E);
  float l = e_src[s * heads + hd] + e_dst[d * heads + hd];
  l = (l > 0.f) ? l : l * NEG_SLOPE;
  logits[t] = l;
  atomicMaxFloat(&m[d * heads + hd], l);
}

// ---------------------------------------------------------------------------
// Edge pass 2: ex = exp(l - m[dst]); store in place; segment atomic-sum denom
// ---------------------------------------------------------------------------
__global__ void edge_exp_sum(const int* __restrict__ srcIdx,
                             const int* __restrict__ dstIdx,
                             int E, int Etot,
                             float* __restrict__ logits,
                             const float* __restrict__ m,
                             float* __restrict__ denom, int heads) {
  int t = blockIdx.x * blockDim.x + threadIdx.x;
  if (t >= Etot * heads) return;
  int e = t / heads;
  int hd = t - e * heads;
  int d = (e < E) ? dstIdx[e] : (e - E);
  float ex = expf(logits[t] - m[d * heads + hd]);
  logits[t] = ex;
  atomicAdd(&denom[d * heads + hd], ex);
}

// ---------------------------------------------------------------------------
// Edge pass 3: alpha = ex / denom[dst]; scatter alpha * H[src] into agg[dst].
// One wave per (edge, head); lanes stride the C channels (coalesced atomics).
// ---------------------------------------------------------------------------
__global__ void edge_scatter(const int* __restrict__ srcIdx,
                             const int* __restrict__ dstIdx,
                             int E, int Etot,
                             const float* __restrict__ ex,
                             const float* __restrict__ denom,
                             const float* __restrict__ H,
                             float* __restrict__ agg,
                             int heads, int C) {
  int lane = threadIdx.x & 31;
  int wid = (int)(((size_t)blockIdx.x * blockDim.x + threadIdx.x) >> 5);
  if (wid >= Etot * heads) return;                // wave-uniform
  int e = wid / heads;
  int hd = wid - e * heads;
  int s = (e < E) ? srcIdx[e] : (e - E);
  int d = (e < E) ? dstIdx[e] : (e - E);
  float alpha = ex[wid] / (denom[d * heads + hd] + 1e-16f);
  const float* hp = H + (size_t)s * heads * C + (size_t)hd * C;
  float* ap = agg + (size_t)d * heads * C + (size_t)hd * C;
  for (int c = lane; c < C; c += 32)
    atomicAdd(&ap[c], alpha * hp[c]);
}

// ---------------------------------------------------------------------------
// out = ELU(agg + b)
// ---------------------------------------------------------------------------
__global__ void bias_elu(const float* __restrict__ agg,
                         const float* __restrict__ b,
                         float* __restrict__ out, int n, int F) {
  int t = blockIdx.x * blockDim.x + threadIdx.x;
  if (t >= n * F) return;
  int f = t % F;
  float v = agg[t] + b[f];
  out[t] = (v > 0.f) ? v : (expf(v) - 1.f);
}

// ---------------------------------------------------------------------------
// d_out[n] = mean_c ELU(agg[n,c] + b[c]); one wave per node, shuffle reduce
// ---------------------------------------------------------------------------
__global__ void final_mean(const float* __restrict__ agg,
                           const float* __restrict__ b,
                           float* __restrict__ out, int n, int F) {
  int lane = threadIdx.x & 31;
  int node = (int)(((size_t)blockIdx.x * blockDim.x + threadIdx.x) >> 5);
  if (node >= n) return;
  const float* ap = agg + (size_t)node * F;
  float sum = 0.f;
  for (int c = lane; c < F; c += 32) {
    float v = ap[c] + b[c];
    v = (v > 0.f) ? v : (expf(v) - 1.f);
    sum += v;
  }
#pragma unroll
  for (int off = 16; off > 0; off >>= 1)
    sum += __shfl_down(sum, off, 32);
  if (lane == 0) out[node] = sum / (float)F;
}

// ---------------------------------------------------------------------------
// Host orchestration
// ---------------------------------------------------------------------------
static inline int cdiv(long long a, long long b) { return (int)((a + b - 1) / b); }

extern "C" void kernel_launch(void* const* d_in, const int* in_sizes, int n_in,
                              void* d_out, int out_size, void* d_ws, size_t ws_size,
                              hipStream_t stream) {
  const float* x      = (const float*)d_in[0];
  const int*   ei     = (const int*)  d_in[1];
  const float* W1     = (const float*)d_in[2];
  const float* a1_src = (const float*)d_in[3];
  const float* a1_dst = (const float*)d_in[4];
  const float* b1     = (const float*)d_in[5];
  const float* W2     = (const float*)d_in[6];
  const float* a2_src = (const float*)d_in[7];
  const float* a2_dst = (const float*)d_in[8];
  const float* b2     = (const float*)d_in[9];
  float* outp = (float*)d_out;

  const int N    = in_sizes[0] / DIN;   // 50000
  const int E    = in_sizes[1] / 2;     // 800000
  const int Etot = E + N;               // with self-loops
  const int F1   = HEADS * HID;         // 256
  const int* srcIdx = ei;
  const int* dstIdx = ei + E;

  // workspace layout (floats)
  float* ws = (float*)d_ws;
  float* h1      = ws;                               // N*256 (reused as ELU out1)
  float* e_src_b = h1 + (size_t)N * F1;              // N*4
  float* e_dst_b = e_src_b + (size_t)N * HEADS;      // N*4
  float* edgebuf = e_dst_b + (size_t)N * HEADS;      // Etot*4
  float* m_b     = edgebuf + (size_t)Etot * HEADS;   // N*4
  float* den_b   = m_b + (size_t)N * HEADS;          // N*4
  float* aggreg  = den_b + (size_t)N * HEADS;        // N*256 (agg1; later h2|agg2)
  float* h2      = aggreg;                           // N*128
  float* agg2    = aggreg + (size_t)N * DOUT;        // N*128

  const int TB = 256;

  // ================= Layer 1 (4 heads, C=64, concat) =================
  {
    int strips = (N / 16) * (F1 / 64);
    gemm_f32_wmma<DIN, F1><<<cdiv(strips, TB / 32), TB, 0, stream>>>(x, W1, h1, N);

    dot_e_kernel<<<cdiv((long long)N * HEADS, TB), TB, 0, stream>>>(
        h1, a1_src, a1_dst, e_src_b, e_dst_b, N, HEADS, HID);

    fill_kernel<<<1024, TB, 0, stream>>>(m_b, -INFINITY, (size_t)N * HEADS);
    fill_kernel<<<1024, TB, 0, stream>>>(den_b, 0.f, (size_t)N * HEADS);
    fill_kernel<<<4096, TB, 0, stream>>>(aggreg, 0.f, (size_t)N * F1);

    int eh = Etot * HEADS;
    edge_logit_max<<<cdiv(eh, TB), TB, 0, stream>>>(srcIdx, dstIdx, E, Etot,
                                                    e_src_b, e_dst_b, edgebuf, m_b, HEADS);
    edge_exp_sum<<<cdiv(eh, TB), TB, 0, stream>>>(srcIdx, dstIdx, E, Etot,
                                                  edgebuf, m_b, den_b, HEADS);
    edge_scatter<<<cdiv((long long)eh * 32, TB), TB, 0, stream>>>(
        srcIdx, dstIdx, E, Etot, edgebuf, den_b, h1, aggreg, HEADS, HID);

    // ELU(agg + b1) -> out1, written over h1 (h1 no longer needed)
    bias_elu<<<cdiv((long long)N * F1, TB), TB, 0, stream>>>(aggreg, b1, h1, N, F1);
  }

  // ================= Layer 2 (1 head, C=128, mean) =================
  {
    int strips = (N / 16) * (DOUT / 64);
    gemm_f32_wmma<F1, DOUT><<<cdiv(strips, TB / 32), TB, 0, stream>>>(h1, W2, h2, N);

    dot_e_kernel<<<cdiv(N, TB), TB, 0, stream>>>(
        h2, a2_src, a2_dst, e_src_b, e_dst_b, N, 1, DOUT);

    fill_kernel<<<1024, TB, 0, stream>>>(m_b, -INFINITY, (size_t)N);
    fill_kernel<<<1024, TB, 0, stream>>>(den_b, 0.f, (size_t)N);
    fill_kernel<<<4096, TB, 0, stream>>>(agg2, 0.f, (size_t)N * DOUT);

    edge_logit_max<<<cdiv(Etot, TB), TB, 0, stream>>>(srcIdx, dstIdx, E, Etot,
                                                      e_src_b, e_dst_b, edgebuf, m_b, 1);
    edge_exp_sum<<<cdiv(Etot, TB), TB, 0, stream>>>(srcIdx, dstIdx, E, Etot,
                                                    edgebuf, m_b, den_b, 1);
    edge_scatter<<<cdiv((long long)Etot * 32, TB), TB, 0, stream>>>(
        srcIdx, dstIdx, E, Etot, edgebuf, den_b, h2, agg2, 1, DOUT);

    // final: d_out[n] = mean_c ELU(agg2[n,c] + b2[c])
    final_mean<<<cdiv((long long)N * 32, TB), TB, 0, stream>>>(agg2, b2, outp, N, DOUT);
  }
}